// EdgeGCN_88914412962544
// MI455X (gfx1250) — compile-verified
//
#include <hip/hip_runtime.h>

#define N_NODES 100000
#define N_EDGES 1600000
#define HID 64

typedef float v2f __attribute__((ext_vector_type(2)));
typedef float v8f __attribute__((ext_vector_type(8)));

// ---------------- degree / normalization ----------------
__global__ void k_init_deg(float* deg) {
    int n = blockIdx.x * blockDim.x + threadIdx.x;
    if (n < N_NODES) deg[n] = 1.0f;               // self-loop
}

__global__ void k_count_deg(const int* __restrict__ ei, float* __restrict__ deg) {
    int e = blockIdx.x * blockDim.x + threadIdx.x;
    if (e < N_EDGES) atomicAdd(&deg[ei[N_EDGES + e]], 1.0f);
}

__global__ void k_fin_dis(float* dis) {
    int n = blockIdx.x * blockDim.x + threadIdx.x;
    if (n < N_NODES) dis[n] = rsqrtf(dis[n]);
}

__global__ void k_zero(float* __restrict__ p, int n) {
    int i = blockIdx.x * blockDim.x + threadIdx.x;
    if (i < n) p[i] = 0.0f;
}

// ---------------- C[M,64] = A[M,K] @ W[K,64] via f32 WMMA ----------------
template <int K>
__global__ void k_gemm64(const float* __restrict__ A, const float* __restrict__ W,
                         float* __restrict__ C, int mtiles) {
    int wid = (int)((blockIdx.x * blockDim.x + threadIdx.x) >> 5);
    if (wid >= mtiles) return;                     // wave-uniform; EXEC stays all-ones
    int lane = threadIdx.x & 31;
    int m16 = lane & 15;
    int hi  = lane >> 4;

    v8f acc0 = {}, acc1 = {}, acc2 = {}, acc3 = {};
    const float* Arow = A + (size_t)(wid * 16 + m16) * K;

    for (int k0 = 0; k0 < K; k0 += 4) {
        int kb = k0 + 2 * hi;                      // A 16x4: lanes 0-15 K={k0,k0+1}, 16-31 K={k0+2,k0+3}
        v2f a; a.x = Arow[kb]; a.y = Arow[kb + 1];
        v2f b0, b1, b2, b3;
        b0.x = W[kb * 64 + m16];        b0.y = W[(kb + 1) * 64 + m16];
        b1.x = W[kb * 64 + 16 + m16];   b1.y = W[(kb + 1) * 64 + 16 + m16];
        b2.x = W[kb * 64 + 32 + m16];   b2.y = W[(kb + 1) * 64 + 32 + m16];
        b3.x = W[kb * 64 + 48 + m16];   b3.y = W[(kb + 1) * 64 + 48 + m16];
        acc0 = __builtin_amdgcn_wmma_f32_16x16x4_f32(false, a, false, b0, (short)0, acc0, false, false);
        acc1 = __builtin_amdgcn_wmma_f32_16x16x4_f32(false, a, false, b1, (short)0, acc1, false, false);
        acc2 = __builtin_amdgcn_wmma_f32_16x16x4_f32(false, a, false, b2, (short)0, acc2, false, false);
        acc3 = __builtin_amdgcn_wmma_f32_16x16x4_f32(false, a, false, b3, (short)0, acc3, false, false);
    }

    float* Cbase = C + (size_t)wid * 16 * 64;
    #pragma unroll
    for (int r = 0; r < 8; ++r) {
        int row = r + 8 * hi;                      // C/D: rows 0-7 lanes 0-15, rows 8-15 lanes 16-31
        Cbase[row * 64 + m16]      = acc0[r];
        Cbase[row * 64 + 16 + m16] = acc1[r];
        Cbase[row * 64 + 32 + m16] = acc2[r];
        Cbase[row * 64 + 48 + m16] = acc3[r];
    }
}

// ---------------- edge scatter-add (one wave per edge, L2-resident) ----------------
__global__ void k_agg(const float* __restrict__ h, const int* __restrict__ ei,
                      const float* __restrict__ dis, float* __restrict__ agg) {
    int w = (int)((blockIdx.x * blockDim.x + threadIdx.x) >> 5);
    if (w >= N_EDGES) return;
    int lane = threadIdx.x & 31;
    int s = ei[w], d = ei[N_EDGES + w];
    float norm = dis[s] * dis[d];
    const float* hs = h + (size_t)s * 64;
    float* ad = agg + (size_t)d * 64;
    atomicAdd(&ad[lane],      hs[lane]      * norm);
    atomicAdd(&ad[lane + 32], hs[lane + 32] * norm);
}

// agg = [relu](agg + t*dis^2 + bias)   (self-loop + bias epilogue, in place)
__global__ void k_post(float* __restrict__ agg, const float* __restrict__ t,
                       const float* __restrict__ dis, const float* __restrict__ b, int relu) {
    int i = blockIdx.x * blockDim.x + threadIdx.x;
    if (i >= N_NODES * 64) return;
    int n = i >> 6, f = i & 63;
    float ds = dis[n];
    float v = agg[i] + t[i] * ds * ds + b[f];
    agg[i] = relu ? fmaxf(v, 0.0f) : v;
}

// h2 = agg + t1*dis^2 + b2 (kept in registers);  a[n]=h2.fW[0:64], bs[n]=h2.fW[64:128]
__global__ void k_node_dots(const float* __restrict__ agg, const float* __restrict__ t1,
                            const float* __restrict__ dis, const float* __restrict__ b2,
                            const float* __restrict__ fW, float* __restrict__ a,
                            float* __restrict__ bs) {
    int w = (int)((blockIdx.x * blockDim.x + threadIdx.x) >> 5);
    if (w >= N_NODES) return;
    int lane = threadIdx.x & 31;
    float d2 = dis[w]; d2 *= d2;
    size_t base = (size_t)w * 64;
    float h0 = agg[base + lane]      + t1[base + lane]      * d2 + b2[lane];
    float h1 = agg[base + lane + 32] + t1[base + lane + 32] * d2 + b2[lane + 32];
    float pa = h0 * fW[lane]      + h1 * fW[lane + 32];
    float pb = h0 * fW[64 + lane] + h1 * fW[96 + lane];
    #pragma unroll
    for (int m = 16; m >= 1; m >>= 1) {
        pa += __shfl_xor(pa, m);
        pb += __shfl_xor(pb, m);
    }
    if (lane == 0) { a[w] = pa; bs[w] = pb; }
}

// u[k] = sum_j eW2[k][j]*fW3[j];   s = eb2.fW3 + fb   (fold eW2/fW through the edge head)
__global__ void k_prep_u(const float* __restrict__ eW2, const float* __restrict__ eb2,
                         const float* __restrict__ fW, const float* __restrict__ fb,
                         float* __restrict__ u, float* __restrict__ sc) {
    int k = threadIdx.x;                           // 64 threads
    const float* f3 = fW + 128;
    float acc = 0.0f;
    for (int j = 0; j < 64; ++j) acc += eW2[k * 64 + j] * f3[j];
    u[k] = acc;
    if (k == 0) {
        float s = fb[0];
        for (int j = 0; j < 64; ++j) s += eb2[j] * f3[j];
        sc[0] = s;
    }
}

// fused edge head: 16 edges/wave; ea@eW1 via WMMA, relu, dot with u, + a[src]+bs[dst]+s
__global__ void k_edge_out(const float* __restrict__ ea, const int* __restrict__ ei,
                           const float* __restrict__ eW1, const float* __restrict__ eb1,
                           const float* __restrict__ u, const float* __restrict__ sc,
                           const float* __restrict__ a, const float* __restrict__ bs,
                           float* __restrict__ out) {
    int w = (int)((blockIdx.x * blockDim.x + threadIdx.x) >> 5);
    if (w >= N_EDGES / 16) return;
    int lane = threadIdx.x & 31;
    int m16 = lane & 15;
    int hi  = lane >> 4;
    int ebase = w * 16;

    const float* Arow = ea + (size_t)(ebase + m16) * 16;   // 16 edges x 16 feats
    v8f acc0 = {}, acc1 = {}, acc2 = {}, acc3 = {};
    #pragma unroll
    for (int ks = 0; ks < 4; ++ks) {
        int kb = ks * 4 + 2 * hi;
        v2f av; av.x = Arow[kb]; av.y = Arow[kb + 1];
        v2f b0, b1, b2, b3;
        b0.x = eW1[kb * 64 + m16];       b0.y = eW1[(kb + 1) * 64 + m16];
        b1.x = eW1[kb * 64 + 16 + m16];  b1.y = eW1[(kb + 1) * 64 + 16 + m16];
        b2.x = eW1[kb * 64 + 32 + m16];  b2.y = eW1[(kb + 1) * 64 + 32 + m16];
        b3.x = eW1[kb * 64 + 48 + m16];  b3.y = eW1[(kb + 1) * 64 + 48 + m16];
        acc0 = __builtin_amdgcn_wmma_f32_16x16x4_f32(false, av, false, b0, (short)0, acc0, false, false);
        acc1 = __builtin_amdgcn_wmma_f32_16x16x4_f32(false, av, false, b1, (short)0, acc1, false, false);
        acc2 = __builtin_amdgcn_wmma_f32_16x16x4_f32(false, av, false, b2, (short)0, acc2, false, false);
        acc3 = __builtin_amdgcn_wmma_f32_16x16x4_f32(false, av, false, b3, (short)0, acc3, false, false);
    }

    float e0 = eb1[m16], e1 = eb1[16 + m16], e2 = eb1[32 + m16], e3 = eb1[48 + m16];
    float u0 = u[m16],   u1 = u[16 + m16],   u2 = u[32 + m16],   u3 = u[48 + m16];

    float res = 0.0f;
    #pragma unroll
    for (int r = 0; r < 8; ++r) {                  // row m = r + 8*hi
        float p = fmaxf(acc0[r] + e0, 0.0f) * u0
                + fmaxf(acc1[r] + e1, 0.0f) * u1
                + fmaxf(acc2[r] + e2, 0.0f) * u2
                + fmaxf(acc3[r] + e3, 0.0f) * u3;
        p += __shfl_xor(p, 1); p += __shfl_xor(p, 2);
        p += __shfl_xor(p, 4); p += __shfl_xor(p, 8);   // reduce over 16 N-lanes
        if (m16 == r) res = p;
    }
    if (m16 < 8) {
        int e = ebase + m16 + 8 * hi;
        int s = ei[e], d = ei[N_EDGES + e];
        out[e] = a[s] + bs[d] + res + sc[0];
    }
}

// ---------------- launch ----------------
extern "C" void kernel_launch(void* const* d_in, const int* in_sizes, int n_in,
                              void* d_out, int out_size, void* d_ws, size_t ws_size,
                              hipStream_t stream) {
    const float* x   = (const float*)d_in[0];
    const int*   ei  = (const int*)d_in[1];     // edge_index [2,E]
    const float* ea  = (const float*)d_in[2];
    const float* W1  = (const float*)d_in[3];
    const float* b1  = (const float*)d_in[4];
    const float* W2  = (const float*)d_in[5];
    const float* b2  = (const float*)d_in[6];
    const float* eW1 = (const float*)d_in[7];
    const float* eb1 = (const float*)d_in[8];
    const float* eW2 = (const float*)d_in[9];
    const float* eb2 = (const float*)d_in[10];
    const float* fW  = (const float*)d_in[11];
    const float* fb  = (const float*)d_in[12];
    float* out = (float*)d_out;

    float* ws  = (float*)d_ws;
    float* t0  = ws;                              // N*64  (t0, later t1)
    float* agg = ws + (size_t)N_NODES * 64;       // N*64  (agg1 -> h in place; reused for agg2)
    float* dis = agg + (size_t)N_NODES * 64;      // N
    float* av  = dis + N_NODES;                   // N
    float* bsv = av + N_NODES;                    // N
    float* u   = bsv + N_NODES;                   // 64
    float* sc  = u + 64;                          // 1

    const int B = 256;
    const int MT = N_NODES / 16;                  // 6250 row tiles

    // degree -> dis = rsqrt(deg)
    k_init_deg<<<(N_NODES + B - 1) / B, B, 0, stream>>>(dis);
    k_count_deg<<<(N_EDGES + B - 1) / B, B, 0, stream>>>(ei, dis);
    k_fin_dis<<<(N_NODES + B - 1) / B, B, 0, stream>>>(dis);

    // layer 1: t0 = x@W1 ; agg = scatter(norm * t0[src]) ; agg = relu(agg + t0/deg + b1)
    k_gemm64<128><<<(MT + 7) / 8, B, 0, stream>>>(x, W1, t0, MT);
    k_zero<<<(N_NODES * 64 + B - 1) / B, B, 0, stream>>>(agg, N_NODES * 64);
    k_agg<<<N_EDGES / 8, B, 0, stream>>>(t0, ei, dis, agg);
    k_post<<<(N_NODES * 64 + B - 1) / B, B, 0, stream>>>(agg, t0, dis, b1, 1);

    // layer 2: t1 = h@W2 (into t0) ; agg2 ; fused epilogue -> per-node dots a,b
    k_gemm64<64><<<(MT + 7) / 8, B, 0, stream>>>(agg, W2, t0, MT);
    k_zero<<<(N_NODES * 64 + B - 1) / B, B, 0, stream>>>(agg, N_NODES * 64);
    k_agg<<<N_EDGES / 8, B, 0, stream>>>(t0, ei, dis, agg);
    k_node_dots<<<N_NODES / 8, B, 0, stream>>>(agg, t0, dis, b2, fW, av, bsv);

    // edge head: fold eW2 & fW, then fused WMMA edge MLP + final score
    k_prep_u<<<1, 64, 0, stream>>>(eW2, eb2, fW, fb, u, sc);
    k_edge_out<<<(N_EDGES / 16) / 8, B, 0, stream>>>(ea, ei, eW1, eb1, u, sc, av, bsv, out);
}